// GIN_NodeWeightEncoder_68719477376
// MI455X (gfx1250) — compile-verified
//
#include <hip/hip_runtime.h>

#define NNODES 50000
#define NEDGES 1600000
#define FDIM 128

typedef __attribute__((ext_vector_type(2))) float v2f;
typedef __attribute__((ext_vector_type(8))) float v8f;

#if defined(__HIP_DEVICE_COMPILE__)
#if !__has_builtin(__builtin_amdgcn_wmma_f32_16x16x4_f32)
#error "gfx1250 fp32 WMMA builtin __builtin_amdgcn_wmma_f32_16x16x4_f32 not available"
#endif
#endif

__device__ __forceinline__ void fadd_global(float* p, float v) {
  // lowers to global_atomic_add_f32 (no CAS loop)
  unsafeAtomicAdd(p, v);
}

// ---------------------------------------------------------------------------
// Scatter-add: one wave (32 lanes) per edge; each lane handles a float4.
// agg must be pre-initialized with the self term (agg = x).
// ---------------------------------------------------------------------------
__global__ __launch_bounds__(256) void scatter_add_kernel(
    const float* __restrict__ feat, const long long* __restrict__ edges,
    float* __restrict__ agg) {
  const int e = (blockIdx.x * 256 + threadIdx.x) >> 5;
  const int lane = threadIdx.x & 31;
  if (e >= NEDGES) return;
  const long long s = edges[e];
  const long long d = edges[NEDGES + e];
  const float4 v = ((const float4*)(feat + s * FDIM))[lane];  // coalesced 512B/wave
  float* dp = agg + d * FDIM + lane * 4;
  fadd_global(dp + 0, v.x);
  fadd_global(dp + 1, v.y);
  fadd_global(dp + 2, v.z);
  fadd_global(dp + 3, v.w);
}

// ---------------------------------------------------------------------------
// 128-wide fp32 GEMM + bias + ReLU using V_WMMA_F32_16X16X4_F32.
// Block = 256 threads = 8 waves; block computes a 128x128 output slab.
// Wave w owns rows [rowBlock + 16w, +16), all 128 columns: 8 v8f accumulators.
//
// LDS layouts:
//   sA: row stride 132 (conflict-free strided reads, 8B-aligned v2f loads)
//   sW: K-interleaved, element (k,n) at sW[(k>>2)*512 + n*4 + (k&3)], so each
//       lane's B fragment (two consecutive K for one N) is ONE aligned
//       ds_load_b64 landing directly in an even VGPR pair -> no mov shuffle.
//       Byte addr = kg*2048 + n*16 + khalf*4: lanes 0-15 banks {4l,4l+1},
//       lanes 16-31 banks {4l+2,4l+3} -> all 64 banks once, conflict-free.
// ---------------------------------------------------------------------------
__global__ __launch_bounds__(256) void gin_gemm128_relu(
    const float* __restrict__ A, const float* __restrict__ W,
    const float* __restrict__ bias, float* __restrict__ Out, int nrows) {
  __shared__ float sA[128 * 132];
  __shared__ float sW[128 * 128];

  const int tid = threadIdx.x;
  const int rowBlock = blockIdx.x * 128;
  const int validRows = nrows - rowBlock;

  // stage W into K-interleaved layout: thread reads float4 (k, n4*4..+3)
  for (int i = tid; i < (128 * 128) / 4; i += 256) {
    const int k = i >> 5, n4 = i & 31;
    const float4 v = ((const float4*)W)[i];
    float* dst = &sW[(k >> 2) * 512 + (k & 3)];
    dst[(n4 * 4 + 0) * 4] = v.x;
    dst[(n4 * 4 + 1) * 4] = v.y;
    dst[(n4 * 4 + 2) * 4] = v.z;
    dst[(n4 * 4 + 3) * 4] = v.w;
  }
  // stage A tile (zero-fill OOB rows)
  for (int i = tid; i < 128 * 32; i += 256) {
    const int r = i >> 5, c4 = i & 31;
    float4 v = make_float4(0.f, 0.f, 0.f, 0.f);
    if (r < validRows) v = ((const float4*)(A + (size_t)(rowBlock + r) * FDIM))[c4];
    *(float4*)&sA[r * 132 + c4 * 4] = v;
  }
  __syncthreads();

  const int wave = tid >> 5;
  const int lane = tid & 31;
  const int waveRow = wave * 16;
  if (rowBlock + waveRow >= nrows) return;  // whole-wave uniform exit (nrows%16==0)

  const int lrow = lane & 15;
  const int khalf = (lane >> 4) << 1;  // lanes 0-15 -> K+0/1, lanes 16-31 -> K+2/3

  v8f c[8];
  const v8f zero = {0.f, 0.f, 0.f, 0.f, 0.f, 0.f, 0.f, 0.f};
#pragma unroll
  for (int t = 0; t < 8; ++t) c[t] = zero;

  const float* aptr = &sA[(waveRow + lrow) * 132 + khalf];
  const float* bptr = &sW[lrow * 4 + khalf];  // + t*64 + kg*512

  for (int kg = 0; kg < 32; ++kg) {          // K groups of 4
    const v2f a = *(const v2f*)(aptr + kg * 4);
    const float* bk = bptr + kg * 512;
#pragma unroll
    for (int t = 0; t < 8; ++t) {
      const v2f b = *(const v2f*)(bk + t * 64);  // one ds_load_b64
      c[t] = __builtin_amdgcn_wmma_f32_16x16x4_f32(
          false, a, false, b, (short)0, c[t], false, false);
    }
  }

  // C/D layout: VGPR r -> M = r (lanes 0-15) or r+8 (lanes 16-31), N = lane%16
  const int mOff = (lane < 16) ? 0 : 8;
#pragma unroll
  for (int t = 0; t < 8; ++t) {
    const int col = t * 16 + lrow;
    const float bv = bias[col];
#pragma unroll
    for (int r = 0; r < 8; ++r) {
      float v = c[t][r] + bv;
      v = v > 0.f ? v : 0.f;
      Out[(size_t)(rowBlock + waveRow + mOff + r) * FDIM + col] = v;
    }
  }
}

// ---------------------------------------------------------------------------
// BatchNorm statistics, F=128. Grid stride (512*256) is a multiple of 128, so
// each thread's feature index is loop-invariant: accumulate in registers,
// finish with 2 global atomics per thread (262K atomics total -> negligible).
// stats[0..127] = sum, stats[128..255] = sumsq.
// ---------------------------------------------------------------------------
__global__ __launch_bounds__(256) void bn_stats128_kernel(
    const float* __restrict__ X, float* __restrict__ stats) {
  const int gtid = blockIdx.x * 256 + threadIdx.x;
  const int f = gtid & (FDIM - 1);
  float s = 0.f, q = 0.f;
  const int total = NNODES * FDIM;
  const int stride = gridDim.x * 256;  // multiple of FDIM by launch config
  for (int idx = gtid; idx < total; idx += stride) {
    const float v = X[idx];
    s += v;
    q += v * v;
  }
  fadd_global(&stats[f], s);
  fadd_global(&stats[FDIM + f], q);
}

// BatchNorm statistics, F=2: register accumulation + wave32 shuffle reduce.
__global__ __launch_bounds__(256) void bn_stats2_kernel(
    const float* __restrict__ X, float* __restrict__ stats) {
  float s0 = 0.f, s1 = 0.f, q0 = 0.f, q1 = 0.f;
  for (int r = blockIdx.x * 256 + threadIdx.x; r < NNODES; r += gridDim.x * 256) {
    const float2 v = ((const float2*)X)[r];
    s0 += v.x; s1 += v.y; q0 += v.x * v.x; q1 += v.y * v.y;
  }
#pragma unroll
  for (int off = 16; off >= 1; off >>= 1) {
    s0 += __shfl_down(s0, off, 32);
    s1 += __shfl_down(s1, off, 32);
    q0 += __shfl_down(q0, off, 32);
    q1 += __shfl_down(q1, off, 32);
  }
  if ((threadIdx.x & 31) == 0) {
    fadd_global(&stats[0], s0);
    fadd_global(&stats[1], s1);
    fadd_global(&stats[2], q0);
    fadd_global(&stats[3], q1);
  }
}

// Apply BN: y = (x - mean) * rsqrt(var + eps) * gamma + beta  (F power of two)
__global__ __launch_bounds__(256) void bn_apply_kernel(
    const float* __restrict__ stats, const float* __restrict__ gamma,
    const float* __restrict__ beta, const float* __restrict__ Xin,
    float* __restrict__ Xout, int F) {
  const int idx = blockIdx.x * 256 + threadIdx.x;
  if (idx >= NNODES * F) return;
  const int f = idx & (F - 1);
  const float invN = 1.0f / (float)NNODES;
  const float mean = stats[f] * invN;
  const float var = stats[F + f] * invN - mean * mean;
  const float inv = rsqrtf(var + 1e-5f);
  Xout[idx] = (Xin[idx] - mean) * inv * gamma[f] + beta[f];
}

// Final head: (50000x128) @ (128x2) + b, ReLU. One wave per row.
__global__ __launch_bounds__(256) void head_kernel(
    const float* __restrict__ A, const float* __restrict__ W22,
    const float* __restrict__ b22, float* __restrict__ Out) {
  const int row = (blockIdx.x * 256 + threadIdx.x) >> 5;
  const int lane = threadIdx.x & 31;
  if (row >= NNODES) return;
  const float4 v = ((const float4*)(A + (size_t)row * FDIM))[lane];
  float c0 = 0.f, c1 = 0.f;
  const float av[4] = {v.x, v.y, v.z, v.w};
#pragma unroll
  for (int j = 0; j < 4; ++j) {
    const int k = lane * 4 + j;
    c0 += av[j] * W22[k * 2 + 0];
    c1 += av[j] * W22[k * 2 + 1];
  }
#pragma unroll
  for (int off = 16; off >= 1; off >>= 1) {
    c0 += __shfl_down(c0, off, 32);
    c1 += __shfl_down(c1, off, 32);
  }
  if (lane == 0) {
    float o0 = c0 + b22[0]; o0 = o0 > 0.f ? o0 : 0.f;
    float o1 = c1 + b22[1]; o1 = o1 > 0.f ? o1 : 0.f;
    Out[row * 2 + 0] = o0;
    Out[row * 2 + 1] = o1;
  }
}

__global__ void zero_kernel(float* p, int n) {
  const int i = blockIdx.x * 256 + threadIdx.x;
  if (i < n) p[i] = 0.f;
}

// ---------------------------------------------------------------------------
extern "C" void kernel_launch(void* const* d_in, const int* in_sizes, int n_in,
                              void* d_out, int out_size, void* d_ws, size_t ws_size,
                              hipStream_t stream) {
  const float* x        = (const float*)d_in[0];
  const long long* edge = (const long long*)d_in[1];   // int64 edge_index (2, E)
  const float* w11 = (const float*)d_in[2];
  const float* b11 = (const float*)d_in[3];
  const float* w12 = (const float*)d_in[4];
  const float* b12 = (const float*)d_in[5];
  const float* g1  = (const float*)d_in[6];
  const float* be1 = (const float*)d_in[7];
  const float* w21 = (const float*)d_in[8];
  const float* b21 = (const float*)d_in[9];
  const float* w22 = (const float*)d_in[10];
  const float* b22 = (const float*)d_in[11];
  const float* g2  = (const float*)d_in[12];
  const float* be2 = (const float*)d_in[13];

  const size_t nodeElems = (size_t)NNODES * FDIM;       // 6.4M floats
  float* bufA  = (float*)d_ws;                          // agg (both layers)
  float* bufB  = bufA + nodeElems;                      // t1 / t3
  float* bufC  = bufB + nodeElems;                      // t2 -> h1 (in place BN)
  float* bufD  = bufC + nodeElems;                      // head output 50000x2
  float* stats = bufD + (size_t)NNODES * 2;             // 256 floats

  const int EDGE_BLOCKS = NEDGES / 8;                   // 8 edges (waves) / block
  const int GEMM_BLOCKS = (NNODES + 127) / 128;         // 391

  // ---- Layer 1 ----
  hipMemcpyAsync(bufA, x, nodeElems * sizeof(float), hipMemcpyDeviceToDevice, stream);
  scatter_add_kernel<<<EDGE_BLOCKS, 256, 0, stream>>>(x, edge, bufA);
  gin_gemm128_relu<<<GEMM_BLOCKS, 256, 0, stream>>>(bufA, w11, b11, bufB, NNODES);
  gin_gemm128_relu<<<GEMM_BLOCKS, 256, 0, stream>>>(bufB, w12, b12, bufC, NNODES);
  zero_kernel<<<1, 256, 0, stream>>>(stats, 256);
  bn_stats128_kernel<<<512, 256, 0, stream>>>(bufC, stats);   // stride % 128 == 0
  bn_apply_kernel<<<(NNODES * FDIM + 255) / 256, 256, 0, stream>>>(
      stats, g1, be1, bufC, bufC, FDIM);

  // ---- Layer 2 ----
  hipMemcpyAsync(bufA, bufC, nodeElems * sizeof(float), hipMemcpyDeviceToDevice, stream);
  scatter_add_kernel<<<EDGE_BLOCKS, 256, 0, stream>>>(bufC, edge, bufA);
  gin_gemm128_relu<<<GEMM_BLOCKS, 256, 0, stream>>>(bufA, w21, b21, bufB, NNODES);
  head_kernel<<<(NNODES + 7) / 8, 256, 0, stream>>>(bufB, w22, b22, bufD);
  zero_kernel<<<1, 256, 0, stream>>>(stats, 256);
  bn_stats2_kernel<<<256, 256, 0, stream>>>(bufD, stats);
  bn_apply_kernel<<<(NNODES * 2 + 255) / 256, 256, 0, stream>>>(
      stats, g2, be2, bufD, (float*)d_out, 2);
}